// LDRLDR2_23476291240734
// MI455X (gfx1250) — compile-verified
//
#include <hip/hip_runtime.h>
#include <hip/hip_bf16.h>

typedef float v2f __attribute__((ext_vector_type(2)));
typedef float v8f __attribute__((ext_vector_type(8)));

#define LSDIM 4096
#define BATCH 64
#define FCDIM 512
#define NCLS  10

// ---------------------------------------------------------------------------
// f32 WMMA: D(16x16) = A(16x4) * B(4x16) + C, exact f32 accumulation.
// 8-arg form: (neg_a, A, neg_b, B, c_mod, C, reuse_a, reuse_b)
// ---------------------------------------------------------------------------
__device__ __forceinline__ v8f wmma_f32x4(v2f a, v2f b, v8f c) {
  return __builtin_amdgcn_wmma_f32_16x16x4_f32(false, a, false, b, (short)0, c,
                                               false, false);
}

// ---------------------------------------------------------------------------
// Kernel 1: cumulative product d[k] = prod_{t<k} subd[t], plus reciprocals.
// One block of 1024 threads, 4 elements each, LDS Hillis-Steele scan (x op).
// ---------------------------------------------------------------------------
__global__ __launch_bounds__(1024) void cumprod_kernel(
    const float* __restrict__ subd, float* __restrict__ d,
    float* __restrict__ rd) {
  __shared__ float tot[1024];
  const int tid = threadIdx.x;
  const int base = tid * 4;
  const float p0 = (base == 0) ? 1.0f : subd[base - 1];
  const float p1 = subd[base + 0];
  const float p2 = subd[base + 1];
  const float p3 = subd[base + 2];
  const float q0 = p0, q1 = q0 * p1, q2 = q1 * p2, q3 = q2 * p3;
  tot[tid] = q3;
  __syncthreads();
  for (int off = 1; off < 1024; off <<= 1) {
    const float u = (tid >= off) ? tot[tid - off] : 1.0f;
    __syncthreads();
    tot[tid] *= u;
    __syncthreads();
  }
  const float e = (tid == 0) ? 1.0f : tot[tid - 1];
  const float d0 = e * q0, d1 = e * q1, d2 = e * q2, d3 = e * q3;
  d[base + 0] = d0;  rd[base + 0] = 1.0f / d0;
  d[base + 1] = d1;  rd[base + 1] = 1.0f / d1;
  d[base + 2] = d2;  rd[base + 2] = 1.0f / d2;
  d[base + 3] = d3;  rd[base + 3] = 1.0f / d3;
}

// ---------------------------------------------------------------------------
// Kernel 2: C[k,l] = rda[k]*rdb[l] * sum_i G[i,k]*H[i,l]   (rank-R GEMM)
// Workgroup (256 thr / 8 waves) computes a 128x128 tile of C via WMMA f32.
// G,H row stride is LSDIM. grid = (rows/128, cols/128).
// ---------------------------------------------------------------------------
template <int R>
__global__ __launch_bounds__(256) void build_c_kernel(
    const float* __restrict__ G, const float* __restrict__ H,
    const float* __restrict__ rda, const float* __restrict__ rdb,
    float* __restrict__ C, int ldc) {
  __shared__ float Gs[R][128];
  __shared__ float Hs[R][128];
  const int tid = threadIdx.x;
  const int k0 = blockIdx.x * 128;
  const int l0 = blockIdx.y * 128;

  for (int idx = tid; idx < R * 128; idx += 256) {
    const int r = idx >> 7, c = idx & 127;
    Gs[r][c] = G[r * LSDIM + k0 + c];
    Hs[r][c] = H[r * LSDIM + l0 + c];
  }
  __syncthreads();

  const int lane = tid & 31;
  const int wv = tid >> 5;      // wave id -> k-subtile row (8x8 subtile grid)
  const int m = lane & 15;
  const int hi = lane >> 4;     // lane-half: K offset {0,1} vs {2,3}
  const int ti = wv;

#pragma unroll
  for (int tj = 0; tj < 8; ++tj) {
    v8f acc = {0, 0, 0, 0, 0, 0, 0, 0};
#pragma unroll
    for (int kk = 0; kk < R; kk += 4) {
      v2f a, b;
      a.x = Gs[kk + 2 * hi + 0][ti * 16 + m];
      a.y = Gs[kk + 2 * hi + 1][ti * 16 + m];
      b.x = Hs[kk + 2 * hi + 0][tj * 16 + m];
      b.y = Hs[kk + 2 * hi + 1][tj * 16 + m];
      acc = wmma_f32x4(a, b, acc);
    }
#pragma unroll
    for (int r = 0; r < 8; ++r) {
      const int gk = k0 + ti * 16 + r + 8 * hi;
      const int gl = l0 + tj * 16 + m;
      C[gk * ldc + gl] = rda[gk] * rdb[gl] * acc[r];
    }
  }
}

// ---------------------------------------------------------------------------
// Kernel 3: in-place diagonal prefix sum  S[k,l] = C[k,l] + S[k-1,l-1],
// epilogue A[k,l] = da[k]*db[l]*S[k,l].
// Bands of 64 diagonals per workgroup (128 threads). Windows staged in LDS
// (pad stride 65 / 129 -> conflict-free diagonal walks), carries in registers.
// grid = NR/64 (lower, d>=0) + NC/64 (upper, d<0) workgroups.
// ---------------------------------------------------------------------------
__global__ __launch_bounds__(128) void diag_scan_kernel(
    float* __restrict__ C, const float* __restrict__ da,
    const float* __restrict__ db, int NR, int NC, int ldc) {
  __shared__ float W[128 * 65];  // lower: 128x64 (stride 65); upper: 64x128 (129)
  const int tid = threadIdx.x;
  const int NRt = NR >> 6, NCt = NC >> 6;

  if ((int)blockIdx.x < NRt) {
    // ---- lower bands: d = 64*g + t, elements (d+j, j) ----
    const int g = blockIdx.x;
    float carry = 0.0f;
    const int smax = min(NCt, NRt - g);
    for (int s = 0; s < smax; ++s) {
      const int row0 = (g + s) << 6, col0 = s << 6;
      for (int idx = tid; idx < 128 * 64; idx += 128) {
        const int r = idx >> 6, c = idx & 63;
        const int gr = row0 + r;
        W[r * 65 + c] = (gr < NR) ? C[gr * ldc + col0 + c] : 0.0f;
      }
      __syncthreads();
      if (tid < 64) {
#pragma unroll 1
        for (int j = 0; j < 64; ++j) {
          const int gr = row0 + tid + j;
          if (gr >= NR) break;
          carry += W[(tid + j) * 65 + j];
          W[(tid + j) * 65 + j] = carry;
        }
      }
      __syncthreads();
      for (int idx = tid; idx < 128 * 64; idx += 128) {
        const int r = idx >> 6, c = idx & 63;
        const int dl = r - c;
        const int gr = row0 + r;
        if (dl >= 0 && dl < 64 && gr < NR)
          C[gr * ldc + col0 + c] = da[gr] * db[col0 + c] * W[r * 65 + c];
      }
      __syncthreads();
    }
  } else {
    // ---- upper bands: d' = 64*D + t (skip d'=0), elements (j, j+d') ----
    const int D = blockIdx.x - NRt;
    float carry = 0.0f;
    const int smax = min(NRt, NCt - D);
    for (int s = 0; s < smax; ++s) {
      const int row0 = s << 6, col0 = (s + D) << 6;
      for (int idx = tid; idx < 64 * 128; idx += 128) {
        const int r = idx >> 7, c = idx & 127;
        const int gc = col0 + c;
        W[r * 129 + c] = (gc < NC) ? C[(row0 + r) * ldc + gc] : 0.0f;
      }
      __syncthreads();
      if (tid < 64 && !(D == 0 && tid == 0)) {
#pragma unroll 1
        for (int j = 0; j < 64; ++j) {
          const int gc = col0 + tid + j;
          if (gc >= NC) break;
          carry += W[j * 129 + tid + j];
          W[j * 129 + tid + j] = carry;
        }
      }
      __syncthreads();
      for (int idx = tid; idx < 64 * 128; idx += 128) {
        const int r = idx >> 7, c = idx & 127;
        const int dl = c - r;
        const int gc = col0 + c;
        if (dl >= 0 && dl < 64 && !(D == 0 && dl == 0) && gc < NC)
          C[(row0 + r) * ldc + gc] = da[row0 + r] * db[gc] * W[r * 129 + c];
      }
      __syncthreads();
    }
  }
}

// ---------------------------------------------------------------------------
// Kernel 4: Y(64 x NF) = act( X(64 x Kw) @ A(NF x Kw)^T + bias ), WMMA f32.
// Workgroup: 64(batch) x 64(features) tile; LDS-tiled K loop (64 per step).
// grid.x = NF/64.
// ---------------------------------------------------------------------------
__global__ __launch_bounds__(256) void apply_kernel(
    const float* __restrict__ X, int ldx, int Kw, const float* __restrict__ A,
    int lda, const float* __restrict__ bias, float* __restrict__ Y, int ldy,
    int relu) {
  __shared__ float Xs[64][65];
  __shared__ float As[64][65];
  const int tid = threadIdx.x;
  const int f0 = blockIdx.x * 64;
  const int lane = tid & 31, wv = tid >> 5;
  const int m = lane & 15, hi = lane >> 4;
  const int mi = wv & 3;            // batch subtile
  const int ni0 = (wv >> 2) * 2;    // feature subtile pair

  v8f acc0 = {0, 0, 0, 0, 0, 0, 0, 0};
  v8f acc1 = {0, 0, 0, 0, 0, 0, 0, 0};

  for (int l0 = 0; l0 < Kw; l0 += 64) {
    for (int idx = tid; idx < 64 * 64; idx += 256) {
      const int r = idx >> 6, c = idx & 63;
      Xs[r][c] = X[r * ldx + l0 + c];
      As[r][c] = A[(f0 + r) * lda + l0 + c];
    }
    if (l0 + 64 < Kw)  // stream-ahead hint for the A panel
      __builtin_prefetch(&A[(f0 + (tid >> 2)) * lda + l0 + 64], 0, 0);
    __syncthreads();
#pragma unroll
    for (int ks = 0; ks < 16; ++ks) {
      const int kk = ks * 4 + 2 * hi;
      v2f a, b0, b1;
      a.x = Xs[mi * 16 + m][kk];
      a.y = Xs[mi * 16 + m][kk + 1];
      b0.x = As[ni0 * 16 + m][kk];
      b0.y = As[ni0 * 16 + m][kk + 1];
      b1.x = As[(ni0 + 1) * 16 + m][kk];
      b1.y = As[(ni0 + 1) * 16 + m][kk + 1];
      acc0 = wmma_f32x4(a, b0, acc0);
      acc1 = wmma_f32x4(a, b1, acc1);
    }
    __syncthreads();
  }

#pragma unroll
  for (int r = 0; r < 8; ++r) {
    const int b = mi * 16 + r + 8 * hi;
    const int f1 = f0 + ni0 * 16 + m;
    const int f2 = f0 + (ni0 + 1) * 16 + m;
    float v1 = acc0[r] + bias[f1];
    float v2 = acc1[r] + bias[f2];
    if (relu) { v1 = fmaxf(v1, 0.0f); v2 = fmaxf(v2, 0.0f); }
    Y[b * ldy + f1] = v1;
    Y[b * ldy + f2] = v2;
  }
}

// ---------------------------------------------------------------------------
// Kernel 5: logits (64x10) = h2(64x512) @ W^T + b  (tiny, plain FMA)
// ---------------------------------------------------------------------------
__global__ __launch_bounds__(256) void logits_kernel(
    const float* __restrict__ h2, const float* __restrict__ Wl,
    const float* __restrict__ bl, float* __restrict__ out) {
  const int tid = threadIdx.x;
  for (int o = tid; o < BATCH * NCLS; o += 256) {
    const int b = o / NCLS, c = o % NCLS;
    const float4* hp = (const float4*)(h2 + b * FCDIM);
    const float4* wp = (const float4*)(Wl + c * FCDIM);
    float s = 0.0f;
    for (int i = 0; i < FCDIM / 4; ++i) {
      const float4 hv = hp[i], wv = wp[i];
      s += hv.x * wv.x + hv.y * wv.y + hv.z * wv.z + hv.w * wv.w;
    }
    out[o] = s + bl[c];
  }
}

// ---------------------------------------------------------------------------
extern "C" void kernel_launch(void* const* d_in, const int* in_sizes, int n_in,
                              void* d_out, int out_size, void* d_ws,
                              size_t ws_size, hipStream_t stream) {
  (void)in_sizes; (void)n_in; (void)out_size; (void)ws_size;
  const float* x   = (const float*)d_in[0];
  const float* G1  = (const float*)d_in[1];
  const float* H1  = (const float*)d_in[2];
  const float* sA1 = (const float*)d_in[3];
  const float* sB1 = (const float*)d_in[4];
  const float* b1  = (const float*)d_in[5];
  const float* G2  = (const float*)d_in[6];
  const float* H2  = (const float*)d_in[7];
  const float* sA2 = (const float*)d_in[8];
  const float* sB2 = (const float*)d_in[9];
  const float* b2  = (const float*)d_in[10];
  const float* Wl  = (const float*)d_in[11];
  const float* bl  = (const float*)d_in[12];
  float* out = (float*)d_out;
  float* ws = (float*)d_ws;

  float* da1  = ws + 0;      float* rda1 = ws + 4096;
  float* db1  = ws + 8192;   float* rdb1 = ws + 12288;
  float* da2  = ws + 16384;  float* rda2 = ws + 20480;
  float* db2  = ws + 24576;  float* rdb2 = ws + 28672;
  float* h1   = ws + 32768;                          // 64*4096
  float* h2   = ws + 32768 + BATCH * LSDIM;          // 64*512
  float* Cbuf = ws + 32768 + BATCH * LSDIM + BATCH * FCDIM;  // <= 4096*1024

  cumprod_kernel<<<1, 1024, 0, stream>>>(sA1, da1, rda1);
  cumprod_kernel<<<1, 1024, 0, stream>>>(sB1, db1, rdb1);
  cumprod_kernel<<<1, 1024, 0, stream>>>(sA2, da2, rda2);
  cumprod_kernel<<<1, 1024, 0, stream>>>(sB2, db2, rdb2);

  // Layer 1 operator: only columns [0,1024) matter (input zero-padded).
  build_c_kernel<48><<<dim3(32, 8), 256, 0, stream>>>(G1, H1, rda1, rdb1,
                                                      Cbuf, 1024);
  diag_scan_kernel<<<64 + 16, 128, 0, stream>>>(Cbuf, da1, db1, 4096, 1024,
                                                1024);
  apply_kernel<<<64, 256, 0, stream>>>(x, 1024, 1024, Cbuf, 1024, b1, h1,
                                       4096, 1);

  // Layer 2 operator: only rows [0,512) matter (classifier reads h2[:, :512]).
  build_c_kernel<16><<<dim3(4, 32), 256, 0, stream>>>(G2, H2, rda2, rdb2,
                                                      Cbuf, 4096);
  diag_scan_kernel<<<8 + 64, 128, 0, stream>>>(Cbuf, da2, db2, 512, 4096,
                                               4096);
  apply_kernel<<<8, 256, 0, stream>>>(h1, 4096, 4096, Cbuf, 4096, b2, h2,
                                      512, 1);

  logits_kernel<<<1, 256, 0, stream>>>(h2, Wl, bl, out);
}